// MyRNN_14955076124800
// MI455X (gfx1250) — compile-verified
//
#include <hip/hip_runtime.h>
#include <math.h>

// Problem constants (match the reference).
#define BB 256
#define TT 4096
#define UU 32

typedef __attribute__((ext_vector_type(2))) float v2f;
typedef __attribute__((ext_vector_type(8))) float v8f;

// gfx1250 has V_TANH_F32 (TRANS op); probe-confirmed it lowers via this builtin.
__device__ __forceinline__ float fast_tanh(float v) {
#if __has_builtin(__builtin_amdgcn_tanhf)
  return __builtin_amdgcn_tanhf(v);
#elif __has_builtin(__builtin_amdgcn_tanh_f32)
  return __builtin_amdgcn_tanh_f32(v);
#else
  return tanhf(v);
#endif
}

// Broadcast lane k of v to the whole wave. Constant k -> v_readlane_b32 into an
// SGPR; consuming FMAs use the SGPR operand directly (no ds_bpermute traffic).
__device__ __forceinline__ float bcast_lane(float v, int k) {
#if __has_builtin(__builtin_amdgcn_readlane)
  return __uint_as_float(__builtin_amdgcn_readlane(__float_as_uint(v), k));
#else
  return __shfl(v, k, 32);
#endif
}

// dot(w, broadcast-vector v) + bias with 4 independent accumulator chains.
__device__ __forceinline__ float matvec32(const float (&w)[32], float v,
                                          float bias) {
  float a0 = 0.0f, a1 = 0.0f, a2 = 0.0f, a3 = 0.0f;
#pragma unroll
  for (int k = 0; k < 32; k += 4) {
    a0 = fmaf(w[k + 0], bcast_lane(v, k + 0), a0);
    a1 = fmaf(w[k + 1], bcast_lane(v, k + 1), a1);
    a2 = fmaf(w[k + 2], bcast_lane(v, k + 2), a2);
    a3 = fmaf(w[k + 3], bcast_lane(v, k + 3), a3);
  }
  return ((a0 + a1) + (a2 + a3)) + bias;
}

// ---------------------------------------------------------------------------
// Phase 1: proj = tanh(x @ W1 + b1) using V_WMMA_F32_16X16X4_F32.
// W1 held in B-layout registers; C initialized to b1; x streamed with NT loads
// so the 128 MB proj output stays resident in the 192 MB L2 for the scan.
// Layouts per ISA §7.12.2 (wave32):
//   A 16x4 f32 : lanes 0-15 -> M=lane, {K=0,K=1}; lanes 16-31 -> M=lane-16, {K=2,K=3}
//   B 4x16 f32 : lanes 0-15 -> N=lane, {K=0,K=1}; lanes 16-31 -> N=lane-16, {K=2,K=3}
//   C/D 16x16  : VGPR r -> (M=r, N=lane) lanes 0-15 ; (M=r+8, N=lane-16) lanes 16-31
// ---------------------------------------------------------------------------
#define PROJ_WAVES 8   // waves per block (256 threads)
#define PROJ_TILES 8   // 16-row tiles per wave -> 128 rows/wave, 1024 rows/block

__global__ __launch_bounds__(256) void proj_wmma_kernel(
    const float* __restrict__ x, const float* __restrict__ W1,
    const float* __restrict__ b1, float* __restrict__ proj) {
  const int lane   = threadIdx.x & 31;
  const int wave   = threadIdx.x >> 5;
  const int half   = lane >> 4;   // 0: lanes 0-15, 1: lanes 16-31
  const int lane16 = lane & 15;

  // W1 in B-matrix layout: Breg[kk][h] covers K = kk*4..kk*4+3, N-half h.
  v2f Breg[8][2];
#pragma unroll
  for (int kk = 0; kk < 8; ++kk) {
#pragma unroll
    for (int h = 0; h < 2; ++h) {
      const int k0 = kk * 4 + half * 2;
      const int n  = h * 16 + lane16;
      v2f bb;
      bb.x = W1[(k0 + 0) * UU + n];
      bb.y = W1[(k0 + 1) * UU + n];
      Breg[kk][h] = bb;
    }
  }
  const float bias0 = b1[lane16];
  const float bias1 = b1[16 + lane16];

  const long long rowBase0 =
      (long long)(blockIdx.x * PROJ_WAVES + wave) * (16LL * PROJ_TILES);

#pragma unroll 1
  for (int tile = 0; tile < PROJ_TILES; ++tile) {
    const long long rowBase = rowBase0 + tile * 16;

    v8f acc0, acc1;
#pragma unroll
    for (int r = 0; r < 8; ++r) { acc0[r] = bias0; acc1[r] = bias1; }

    // A-tile base for this lane: row = rowBase + lane16, K offset half*2.
    const float* arow = x + (rowBase + lane16) * UU + half * 2;
#pragma unroll
    for (int kk = 0; kk < 8; ++kk) {
      // 8-byte aligned non-temporal load of {x[row,k], x[row,k+1]}.
      v2f A = __builtin_nontemporal_load(
          reinterpret_cast<const v2f*>(arow + kk * 4));
      acc0 = __builtin_amdgcn_wmma_f32_16x16x4_f32(
          false, A, false, Breg[kk][0], (short)0, acc0, false, false);
      acc1 = __builtin_amdgcn_wmma_f32_16x16x4_f32(
          false, A, false, Breg[kk][1], (short)0, acc1, false, false);
    }

    // tanh + store (regular temporal hint: keep proj hot in L2).
#pragma unroll
    for (int r = 0; r < 8; ++r) {
      const long long row = rowBase + r + half * 8;
      proj[row * UU + lane16]      = fast_tanh(acc0[r]);
      proj[row * UU + 16 + lane16] = fast_tanh(acc1[r]);
    }
  }
}

// ---------------------------------------------------------------------------
// Phase 2: sequential scan + mean pool + Dense(1).
// One wave per batch element (256 waves). Lane j owns column j of W2 in
// registers; state[k] broadcast via v_readlane into SGPRs (FMA consumes the
// SGPR operand). Serial chain per step: ~8 FMAs + tree add + v_tanh_f32.
// proj rows come from L2; the masked prefetch + pipelined next-load hide the
// rest.
// ---------------------------------------------------------------------------
__global__ __launch_bounds__(256) void rnn_scan_kernel(
    const float* __restrict__ proj, const float* __restrict__ W2,
    const float* __restrict__ b2, const float* __restrict__ Wc,
    const float* __restrict__ bc, float* __restrict__ out) {
  const int lane = threadIdx.x & 31;
  const int b    = blockIdx.x * (256 / 32) + (threadIdx.x >> 5);

  float w[32];
#pragma unroll
  for (int k = 0; k < 32; ++k) w[k] = W2[k * UU + lane];
  const float bias = b2[lane];

  const float* __restrict__ p = proj + (long long)b * TT * UU;

  float state = 0.0f;
  float sum   = 0.0f;
  float px    = p[lane];

#pragma unroll 1
  for (int t = 0; t < TT - 1; ++t) {
    // Branchless, always-in-bounds prefetch (TT is a power of two).
    __builtin_prefetch(&p[((t + 8) & (TT - 1)) * UU + lane], 0, 3);
    const float px_next = p[(t + 1) * UU + lane];

    state = px + fast_tanh(matvec32(w, state, bias));
    sum  += state;
    px    = px_next;
  }
  // Final step (t = TT-1), no next-load.
  state = px + fast_tanh(matvec32(w, state, bias));
  sum  += state;

  // pooled[j] = sum/T ; out[b] = sum_j pooled[j]*Wc[j] + bc
  float contrib = (sum * (1.0f / (float)TT)) * Wc[lane];
#pragma unroll
  for (int off = 16; off > 0; off >>= 1)
    contrib += __shfl_xor(contrib, off, 32);
  if (lane == 0) out[b] = contrib + bc[0];
}

// ---------------------------------------------------------------------------
// Fallback: fully fused single pass (used only if d_ws < 128 MB). Computes
// px_t on the fly; the px dot-product is off the serial chain so it overlaps
// the scan's dependence chain.
// ---------------------------------------------------------------------------
__global__ __launch_bounds__(256) void rnn_fused_kernel(
    const float* __restrict__ x, const float* __restrict__ W1,
    const float* __restrict__ b1, const float* __restrict__ W2,
    const float* __restrict__ b2, const float* __restrict__ Wc,
    const float* __restrict__ bc, float* __restrict__ out) {
  const int lane = threadIdx.x & 31;
  const int b    = blockIdx.x * (256 / 32) + (threadIdx.x >> 5);

  float w1c[32], w2c[32];
#pragma unroll
  for (int k = 0; k < 32; ++k) {
    w1c[k] = W1[k * UU + lane];
    w2c[k] = W2[k * UU + lane];
  }
  const float bias1 = b1[lane];
  const float bias2 = b2[lane];

  const float* __restrict__ xb = x + (long long)b * TT * UU;

  float state = 0.0f;
  float sum   = 0.0f;
  float xv    = __builtin_nontemporal_load(&xb[lane]);

#pragma unroll 1
  for (int t = 0; t < TT - 1; ++t) {
    const float xv_next =
        __builtin_nontemporal_load(&xb[(t + 1) * UU + lane]);

    const float px = fast_tanh(matvec32(w1c, xv, bias1));
    state = px + fast_tanh(matvec32(w2c, state, bias2));
    sum  += state;
    xv    = xv_next;
  }
  {
    const float px = fast_tanh(matvec32(w1c, xv, bias1));
    state = px + fast_tanh(matvec32(w2c, state, bias2));
    sum  += state;
  }

  float contrib = (sum * (1.0f / (float)TT)) * Wc[lane];
#pragma unroll
  for (int off = 16; off > 0; off >>= 1)
    contrib += __shfl_xor(contrib, off, 32);
  if (lane == 0) out[b] = contrib + bc[0];
}

// ---------------------------------------------------------------------------
extern "C" void kernel_launch(void* const* d_in, const int* in_sizes, int n_in,
                              void* d_out, int out_size, void* d_ws,
                              size_t ws_size, hipStream_t stream) {
  const float* x  = (const float*)d_in[0];
  const float* W1 = (const float*)d_in[1];
  const float* b1 = (const float*)d_in[2];
  const float* W2 = (const float*)d_in[3];
  const float* b2 = (const float*)d_in[4];
  const float* Wc = (const float*)d_in[5];
  const float* bc = (const float*)d_in[6];
  float* out = (float*)d_out;

  const size_t need = (size_t)BB * TT * UU * sizeof(float);  // 128 MB
  if (ws_size >= need) {
    float* proj = (float*)d_ws;
    const long long rows = (long long)BB * TT;                    // 1,048,576
    const int rows_per_block = PROJ_WAVES * 16 * PROJ_TILES;      // 1024
    proj_wmma_kernel<<<(int)(rows / rows_per_block), 256, 0, stream>>>(
        x, W1, b1, proj);
    rnn_scan_kernel<<<BB / 8, 256, 0, stream>>>(proj, W2, b2, Wc, bc, out);
  } else {
    rnn_fused_kernel<<<BB / 8, 256, 0, stream>>>(x, W1, b1, W2, b2, Wc, bc,
                                                 out);
  }
}